// NNconv_55731495632943
// MI455X (gfx1250) — compile-verified
//
#include <hip/hip_runtime.h>

// ---------------------------------------------------------------------------
// NNConv fused kernel for gfx1250 (MI455X).
//
// msg[e,o] = sum_{i,k} x[e,i]*h[e,k]*W2[k,16i+o] + sum_i x[e,i]*b2[16i+o]
//  => GEMM:  P_aug (E x 272) @ W2' (272 x 16), K-tiled with
//     V_WMMA_F32_16X16X4_F32 (exact f32, matches reference precision).
// W2' is pre-swizzled ONCE into d_ws in WMMA B-fragment order by a prep
// kernel; each block copies it to LDS (ds_load_b64 per K-tile, conflict-free).
// Scatter-add via global_atomic_add_f32. Root term = separate WMMA kernel
// that also initializes d_out.
// ---------------------------------------------------------------------------

#define C_IN      16
#define C_OUT     16
#define EDGE_DIM  8
#define KT        68            // 272 / 4 K-tiles (64 for W2, 4 for b2 part)
#define WPB       8             // waves per block (wave32)
#define BLOCK_THREADS 256
#define GROUPS_PER_WAVE 4       // 32-edge groups per wave (setup amortization)

typedef float v2f __attribute__((ext_vector_type(2)));
typedef float v8f __attribute__((ext_vector_type(8)));

__device__ __forceinline__ v8f wmma4(v2f a, v2f b, v8f c) {
  // (neg_a, A, neg_b, B, c_mod, C, reuse_a, reuse_b)
  return __builtin_amdgcn_wmma_f32_16x16x4_f32(false, a, false, b, (short)0, c,
                                               false, false);
}

// Exact bitwise blend: mask==~0 -> b, mask==0 -> a. Lowers to a single
// v_cndmask/v_bfi and cannot be re-folded into a variable-index register-array
// extract (which previously expanded into large v_cmp/v_cndmask trees).
__device__ __forceinline__ float lane_sel(float a, float b, unsigned mask) {
  unsigned ua = __float_as_uint(a);
  unsigned ub = __float_as_uint(b);
  return __uint_as_float((ub & mask) | (ua & ~mask));
}

// B-fragment value for augmented K index kk (0..271), output column n.
__device__ __forceinline__ float w2aug(const float* __restrict__ W2g,
                                       const float* __restrict__ b2g,
                                       int kk, int n) {
  if (kk < 256) {                       // W2 part: kk = 16*i + k
    int i = kk >> 4, k = kk & 15;
    return W2g[k * 256 + i * 16 + n];
  }
  return b2g[(kk - 256) * 16 + n];      // b2 part: rows are b2[16i+o]
}

// ---------------------------------------------------------------------------
// Prep kernel: write W2' (augmented, B-fragment order) into workspace.
// wsB[t*32 + lane] = { value(K=4t+2*half, col n), value(K=4t+1+2*half, col n) }
// ---------------------------------------------------------------------------
__global__ __launch_bounds__(BLOCK_THREADS)
void nnconv_prep_kernel(const float* __restrict__ W2g,
                        const float* __restrict__ b2g,
                        float2* __restrict__ wsB) {
  int p = blockIdx.x * BLOCK_THREADS + threadIdx.x;
  if (p >= KT * 32) return;
  int t = p >> 5; int ln = p & 31;
  int bh = ln >> 4; int n = ln & 15;
  float2 v;
  v.x = w2aug(W2g, b2g, 4 * t + 0 + 2 * bh, n);
  v.y = w2aug(W2g, b2g, 4 * t + 1 + 2 * bh, n);
  wsB[p] = v;
}

// ---------------------------------------------------------------------------
// Kernel A: out[n,o] = nf[n,:] @ root + bias[o]   (initializes d_out)
// One wave handles a 16-node tile: 4 x WMMA f32 16x16x4 (K=16).
// ---------------------------------------------------------------------------
__global__ __launch_bounds__(BLOCK_THREADS)
void nnconv_root_kernel(const float* __restrict__ nf,
                        const float* __restrict__ root,
                        const float* __restrict__ bias,
                        float* __restrict__ out, int nTiles, int N) {
  const int wave = threadIdx.x >> 5;
  const int lane = threadIdx.x & 31;
  const int tile = blockIdx.x * WPB + wave;
  if (tile >= nTiles) return;                 // wave-uniform exit
  const int m    = lane & 15;
  const int half = lane >> 4;
  const unsigned msk = 0u - (unsigned)half;   // all-ones on upper half-wave
  const int n0   = tile * 16;

  int nn = n0 + m; if (nn >= N) nn = N - 1;   // clamp (N%16==0 normally)
  const float4* xp = reinterpret_cast<const float4*>(nf + (size_t)nn * C_IN);
  float4 q0 = xp[0], q1 = xp[1], q2 = xp[2], q3 = xp[3];
  float x[16] = {q0.x,q0.y,q0.z,q0.w, q1.x,q1.y,q1.z,q1.w,
                 q2.x,q2.y,q2.z,q2.w, q3.x,q3.y,q3.z,q3.w};

  // per-lane K selection: A element (VGPR j, tile t) needs x[4t+j+2*half]
  float xs[8];
#pragma unroll
  for (int p = 0; p < 8; ++p) {
    const int c = 4 * (p >> 1) + (p & 1);
    xs[p] = lane_sel(x[c], x[c + 2], msk);
  }

  v8f acc;
  float bv = bias[m];
#pragma unroll
  for (int r = 0; r < 8; ++r) acc[r] = bv;

#pragma unroll
  for (int t = 0; t < 4; ++t) {
    int kbase = 4 * t + 2 * half;             // runtime in address only
    v2f b; b.x = root[(kbase + 0) * C_OUT + m];
           b.y = root[(kbase + 1) * C_OUT + m];
    v2f a; a.x = xs[2 * t]; a.y = xs[2 * t + 1];
    acc = wmma4(a, b, acc);
  }

  if (n0 + 16 <= N) {                         // fast path: unguarded stores
#pragma unroll
    for (int r = 0; r < 8; ++r)
      out[(size_t)(n0 + r + 8 * half) * C_OUT + m] = acc[r];
  } else {
#pragma unroll
    for (int r = 0; r < 8; ++r) {
      int row = n0 + r + 8 * half;
      if (row < N) out[(size_t)row * C_OUT + m] = acc[r];
    }
  }
}

// ---------------------------------------------------------------------------
// Kernel B: fused edge-MLP + per-edge matmul + scatter-add.
// Each wave handles GROUPS_PER_WAVE groups of 32 edges
// (2 WMMA M-subtiles per group, sharing the LDS B fragments).
// ---------------------------------------------------------------------------
__global__ __launch_bounds__(BLOCK_THREADS)
void nnconv_edge_kernel(const float* __restrict__ nf,
                        const unsigned int* __restrict__ eidx, // int64 as u32 pairs
                        const float* __restrict__ eattr,
                        const float* __restrict__ W1g,
                        const float* __restrict__ b1g,
                        const float* __restrict__ W2g,
                        const float* __restrict__ b2g,
                        const float2* __restrict__ wsB,        // may be null
                        float* __restrict__ out, int E) {
  __shared__ float2 sB[KT * 32];                  // W2' in B-fragment order
  __shared__ float  sW1[EDGE_DIM * C_OUT];
  __shared__ float  sB1[C_OUT];

  const int tid = threadIdx.x;

  if (wsB != nullptr) {
    // straight vectorized copy of the pre-swizzled fragments (17 KB)
    const float4* s4 = reinterpret_cast<const float4*>(wsB);
    float4*       d4 = reinterpret_cast<float4*>(sB);
    for (int i = tid; i < (KT * 32) / 2; i += BLOCK_THREADS) d4[i] = s4[i];
  } else {
    // fallback: build fragments in-block
    for (int p = tid; p < KT * 32; p += BLOCK_THREADS) {
      int t = p >> 5; int ln = p & 31;
      int bh = ln >> 4; int n = ln & 15;
      float2 v;
      v.x = w2aug(W2g, b2g, 4 * t + 0 + 2 * bh, n);
      v.y = w2aug(W2g, b2g, 4 * t + 1 + 2 * bh, n);
      sB[p] = v;
    }
  }
  if (tid < EDGE_DIM * C_OUT) sW1[tid] = W1g[tid];
  if (tid < C_OUT)            sB1[tid] = b1g[tid];
  __syncthreads();

  const int wave = tid >> 5;
  const int lane = tid & 31;
  const int m    = lane & 15;
  const int half = lane >> 4;
  const unsigned msk = 0u - (unsigned)half;       // all-ones on upper half-wave

  const int nGroups     = (E + 31) / 32;
  const int groupStride = gridDim.x * WPB;

  // wave-uniform grid-stride loop (no barriers inside)
  for (int group = blockIdx.x * WPB + wave; group < nGroups;
       group += groupStride) {
    const int e0 = group * 32;

    // dst index of edge e0+lane (low word of int64); broadcast later via shfl
    unsigned int dstv;
    {
      int el  = e0 + lane;
      int elc = el < E ? el : E - 1;
      dstv = eidx[2 * ((long long)E + elc)];
    }

    float x[2][16], hs[2][8], xs[2][8];
#pragma unroll
    for (int s = 0; s < 2; ++s) {
      int ea = e0 + 16 * s + m;
      if (ea >= E) ea = E - 1;
      // edge attributes (8 f32 = 2x b128)
      const float4* ap =
          reinterpret_cast<const float4*>(eattr + (size_t)ea * EDGE_DIM);
      float4 a0 = ap[0], a1 = ap[1];
      float at[8] = {a0.x,a0.y,a0.z,a0.w, a1.x,a1.y,a1.z,a1.w};
      // gather source node features (64B row = 4x b128)
      unsigned int src = eidx[2 * (long long)ea];
      const float4* xp =
          reinterpret_cast<const float4*>(nf + (size_t)src * C_IN);
      float4 q0 = xp[0], q1 = xp[1], q2 = xp[2], q3 = xp[3];
      x[s][0]=q0.x;  x[s][1]=q0.y;  x[s][2]=q0.z;  x[s][3]=q0.w;
      x[s][4]=q1.x;  x[s][5]=q1.y;  x[s][6]=q1.z;  x[s][7]=q1.w;
      x[s][8]=q2.x;  x[s][9]=q2.y;  x[s][10]=q2.z; x[s][11]=q2.w;
      x[s][12]=q3.x; x[s][13]=q3.y; x[s][14]=q3.z; x[s][15]=q3.w;
      // h = relu(attr @ W1 + b1)
      float h[16];
#pragma unroll
      for (int k = 0; k < 16; ++k) {
        float acc = sB1[k];
#pragma unroll
        for (int a = 0; a < 8; ++a) acc += at[a] * sW1[a * 16 + k];
        h[k] = acc > 0.f ? acc : 0.f;
      }
      // Pre-resolve the lane-half-dependent K offset (+2*half) once, with an
      // exact bit-blend (single VALU op, no dynamic register-array indexing).
#pragma unroll
      for (int p = 0; p < 8; ++p) {
        const int c = 4 * (p >> 1) + (p & 1);
        hs[s][p] = lane_sel(h[c],    h[c + 2],    msk);
        xs[s][p] = lane_sel(x[s][c], x[s][c + 2], msk);
      }
    }

    v8f acc[2];
#pragma unroll
    for (int r = 0; r < 8; ++r) { acc[0][r] = 0.f; acc[1][r] = 0.f; }

    // ---- K loop: P_aug(16x272) @ W2'(272x16), 68 tiles of K=4 ----
#pragma unroll
    for (int t = 0; t < KT; ++t) {
      float2 bb = sB[t * 32 + lane];             // ds_load_b64, conflict-free
      v2f bv; bv.x = bb.x; bv.y = bb.y;
      v2f a0v, a1v;
      if (t < 64) {
        const int i = t >> 2;                    // compile-time after unroll
        const int p = (t & 3) << 1;
        a0v.x = x[0][i] * hs[0][p];  a0v.y = x[0][i] * hs[0][p + 1];
        a1v.x = x[1][i] * hs[1][p];  a1v.y = x[1][i] * hs[1][p + 1];
      } else {                                   // b2 rows: A element = x[i]
        const int p = (t & 3) << 1;
        a0v.x = xs[0][p];  a0v.y = xs[0][p + 1];
        a1v.x = xs[1][p];  a1v.y = xs[1][p + 1];
      }
      acc[0] = wmma4(a0v, bv, acc[0]);
      acc[1] = wmma4(a1v, bv, acc[1]);
    }

    // ---- scatter-add: C element r is msg[edge 16s + r + 8*half][col m] ----
    if (e0 + 32 <= E) {                          // fast path (E%32==0 here)
#pragma unroll
      for (int s = 0; s < 2; ++s) {
#pragma unroll
        for (int r = 0; r < 8; ++r) {
          int eloc = 16 * s + r + 8 * half;
          unsigned int d = (unsigned int)__shfl((int)dstv, eloc, 32);
          unsafeAtomicAdd(out + (size_t)d * C_OUT + m, acc[s][r]);
        }
      }
    } else {
#pragma unroll
      for (int s = 0; s < 2; ++s) {
#pragma unroll
        for (int r = 0; r < 8; ++r) {
          int eloc = 16 * s + r + 8 * half;
          if (e0 + eloc < E) {
            unsigned int d = (unsigned int)__shfl((int)dstv, eloc, 32);
            unsafeAtomicAdd(out + (size_t)d * C_OUT + m, acc[s][r]);
          }
        }
      }
    }
  }
}

// ---------------------------------------------------------------------------
extern "C" void kernel_launch(void* const* d_in, const int* in_sizes, int n_in,
                              void* d_out, int out_size, void* d_ws, size_t ws_size,
                              hipStream_t stream) {
  const float*        nf    = (const float*)d_in[0];
  const unsigned int* eidx  = (const unsigned int*)d_in[1]; // int64 -> u32 pairs
  const float*        eattr = (const float*)d_in[2];
  const float*        W1    = (const float*)d_in[3];
  const float*        b1    = (const float*)d_in[4];
  const float*        W2    = (const float*)d_in[5];
  const float*        b2    = (const float*)d_in[6];
  const float*        root  = (const float*)d_in[7];
  const float*        bias  = (const float*)d_in[8];
  float*              out   = (float*)d_out;

  const int N = in_sizes[0] / C_IN;
  const int E = in_sizes[2] / EDGE_DIM;

  // Workspace-resident pre-swizzled B fragments (17 KB) if scratch allows.
  float2* wsB = nullptr;
  const size_t wsNeed = (size_t)(KT * 32) * sizeof(float2);
  if (d_ws != nullptr && ws_size >= wsNeed) {
    wsB = (float2*)d_ws;
    const int prepBlocks = (KT * 32 + BLOCK_THREADS - 1) / BLOCK_THREADS;
    nnconv_prep_kernel<<<prepBlocks, BLOCK_THREADS, 0, stream>>>(W2, b2, wsB);
  }

  // Kernel A: initializes d_out with root term + bias.
  const int nTiles = (N + 15) / 16;
  const int rootBlocks = (nTiles + WPB - 1) / WPB;
  nnconv_root_kernel<<<rootBlocks, BLOCK_THREADS, 0, stream>>>(
      nf, root, bias, out, nTiles, N);

  // Kernel B: fused edge pipeline; each wave covers GROUPS_PER_WAVE groups.
  const int nGroups = (E + 31) / 32;
  const int edgeBlocks =
      (nGroups + WPB * GROUPS_PER_WAVE - 1) / (WPB * GROUPS_PER_WAVE);
  nnconv_edge_kernel<<<edgeBlocks, BLOCK_THREADS, 0, stream>>>(
      nf, eidx, eattr, W1, b1, W2, b2, wsB, out, E);
}